// LogVoxelizer_13941463843129
// MI455X (gfx1250) — compile-verified
//
#include <hip/hip_runtime.h>
#include <math.h>

// ---------------------------------------------------------------------------
// LogVoxelizer for MI455X (gfx1250).
// Pure scatter problem: ~84 MB total traffic -> HBM-bound (~3.6us at 23.3TB/s).
// No matmul FLOPs => WMMA not applicable. CDNA5-specific path used instead:
// async global->LDS staging of the bin table (ASYNCcnt) + global_prefetch_b8.
// ---------------------------------------------------------------------------

namespace {
constexpr int   kNumX    = 320;
constexpr int   kNumAng  = 192;
constexpr int   kZDepth  = 100;                       // (18 - (-2)) / 0.2
constexpr int   kS       = 3;
constexpr int   kN       = 200000;
constexpr int   kPts     = kS * kN;                   // 600,000 (batch 0 only)
constexpr int   kOutElems = kS * kZDepth * kNumAng * kNumX;   // 18,432,000
constexpr float kLogMin  = 0.99325177301028340f;      // ln(2.7)
constexpr float kLogMax  = 5.10594547390064700f;      // ln(165)
constexpr float kAngle   = 0.43679632679489654f;      // pi/2 - 2.268/2
constexpr float kZMin    = -2.0f;
constexpr float kZStep   = 0.2f;
}  // namespace

typedef int v4i __attribute__((ext_vector_type(4)));
typedef __attribute__((address_space(1))) v4i* glob_v4i_ptr;
typedef __attribute__((address_space(3))) v4i* lds_v4i_ptr;

// Mirror jnp.exp(jnp.linspace(log(xmin), log(xmax), 320)) in f32.
__device__ __forceinline__ float bin_value(int i) {
  const float step = (kLogMax - kLogMin) / (float)(kNumX - 1);
  return expf(fmaf((float)i, step, kLogMin));
}

__global__ void gen_bins_kernel(float* __restrict__ bins) {
  int i = threadIdx.x;
  if (i < kNumX) bins[i] = bin_value(i);
}

__global__ void zero_out_kernel(float4* __restrict__ out, int n4) {
  int i = blockIdx.x * blockDim.x + threadIdx.x;
  if (i < n4) out[i] = make_float4(0.f, 0.f, 0.f, 0.f);
}

__global__ void __launch_bounds__(256) voxelize_kernel(
    const float* __restrict__ pts,   // [S, N, 3] (batch 0 slice)
    const float* __restrict__ bins_g,
    float* __restrict__ out) {       // [S*Z_DEPTH, NUM_ANG, NUM_X]
  __shared__ float s_bins[kNumX];
  const int tid = threadIdx.x;

  // ---- Stage bin table into LDS ------------------------------------------
  if (bins_g != nullptr) {
#if defined(__gfx1250__) && __has_builtin(__builtin_amdgcn_global_load_async_to_lds_b128)
    // CDNA5 async global->LDS copy: 80 lanes x 16B = 1280B, ASYNCcnt-tracked.
    if (tid < kNumX / 4) {
      glob_v4i_ptr gsrc = (glob_v4i_ptr)(bins_g + tid * 4);   // drops const
      lds_v4i_ptr  ldst = (lds_v4i_ptr)(&s_bins[tid * 4]);
      __builtin_amdgcn_global_load_async_to_lds_b128(gsrc, ldst, 0, 0);
    }
#if __has_builtin(__builtin_amdgcn_s_wait_asynccnt)
    __builtin_amdgcn_s_wait_asynccnt(0);
#else
    asm volatile("s_wait_asynccnt 0" ::: "memory");
#endif
#else
    for (int j = tid; j < kNumX; j += (int)blockDim.x) s_bins[j] = bins_g[j];
#endif
  } else {
    // Workspace unavailable: regenerate table in-register (tiny cost).
    for (int j = tid; j < kNumX; j += (int)blockDim.x) s_bins[j] = bin_value(j);
  }
  __syncthreads();

  // ---- One point per thread ----------------------------------------------
  const int i = blockIdx.x * blockDim.x + tid;
  if (i >= kPts) return;

  const float* __restrict__ p = pts + (size_t)i * 3;
  __builtin_prefetch((const void*)p, 0, 1);   // -> global_prefetch_b8
  const float x = p[0];
  const float y = p[1];
  const float z = p[2];

  // searchsorted(X_BINS, x, side='left'): first idx with bins[idx] >= x.
  int lo = 0, hi = kNumX;
  while (lo < hi) {
    int mid = (lo + hi) >> 1;
    if (s_bins[mid] < x) lo = mid + 1; else hi = mid;
  }
  int x_idx = lo < (kNumX - 1) ? lo : (kNumX - 1);

  const float edge   = s_bins[x_idx] / tanf(kAngle);
  const float yscale = (float)kNumAng / (2.0f * edge);
  int y_idx = (int)floorf((y + edge) * yscale);
  int z_idx = (int)floorf((z - kZMin) / kZStep);

  // Inputs are in-bounds by construction; clamp defensively (JAX drops OOB).
  y_idx = y_idx < 0 ? 0 : (y_idx > kNumAng - 1 ? kNumAng - 1 : y_idx);
  z_idx = z_idx < 0 ? 0 : (z_idx > kZDepth - 1 ? kZDepth - 1 : z_idx);

  const int s = i / kN;
  const size_t off =
      ((size_t)(s * kZDepth + z_idx) * kNumAng + (size_t)y_idx) * kNumX +
      (size_t)x_idx;
  out[off] = 1.0f;   // idempotent: duplicate writers all store 1.0
}

extern "C" void kernel_launch(void* const* d_in, const int* in_sizes, int n_in,
                              void* d_out, int out_size, void* d_ws, size_t ws_size,
                              hipStream_t stream) {
  (void)in_sizes; (void)n_in; (void)out_size;
  const float* pts = (const float*)d_in[0];
  float* out = (float*)d_out;

  float* bins = (ws_size >= (size_t)kNumX * sizeof(float)) ? (float*)d_ws : nullptr;
  if (bins) {
    gen_bins_kernel<<<1, kNumX, 0, stream>>>(bins);
  }

  const int n4 = kOutElems / 4;                       // 4,608,000 float4
  zero_out_kernel<<<(n4 + 255) / 256, 256, 0, stream>>>((float4*)out, n4);

  voxelize_kernel<<<(kPts + 255) / 256, 256, 0, stream>>>(pts, bins, out);
}